// DecoderAttentionGRU_12962211299476
// MI455X (gfx1250) — compile-verified
//
#include <hip/hip_runtime.h>
#include <math.h>

typedef __attribute__((ext_vector_type(2))) float v2f;
typedef __attribute__((ext_vector_type(8))) float v8f;

constexpr int V   = 50257;
constexpr int E   = 512;
constexpr int H   = 1024;
constexpr int B   = 64;
constexpr int S   = 512;
constexpr int H2  = 2 * H;      // 2048
constexpr int H3  = 3 * H;      // 3072
constexpr int KIN = H2 + E;     // 2560 = 2H + E

// ---------------------------------------------------------------------------
// Kernel 1: rnn_input[b, 0:2H]   = sum_s enc[s, b, :]   (attn weights are all 1)
//           rnn_input[b, 2H:..]  = emb_table[tok[b], :]
// ---------------------------------------------------------------------------
__global__ void build_rnn_input_kernel(const float* __restrict__ enc,
                                       const int*   __restrict__ tok,
                                       const float* __restrict__ embt,
                                       float*       __restrict__ rnn) {
  int idx = blockIdx.x * blockDim.x + threadIdx.x;
  if (idx >= B * KIN) return;
  int b = idx / KIN;
  int d = idx - b * KIN;
  float val;
  if (d < H2) {
    const float* p = enc + (size_t)b * H2 + d;
    float s = 0.f;
#pragma unroll 4
    for (int si = 0; si < S; ++si) s += p[(size_t)si * (B * H2)];
    val = s;
  } else {
    val = embt[(size_t)tok[b] * E + (d - H2)];
  }
  rnn[idx] = val;
}

// ---------------------------------------------------------------------------
// Shared WMMA f32 GEMM tile: one wave computes D[0:64, n0:n0+16] where
//   D[m][n] = sum_k A[m][k] * W[n][k] + bias[n]
// A: (64, K) row-major, lda; W: (Nout, K) row-major (i.e. B = W^T).
// Uses V_WMMA_F32_16X16X4_F32; fragment layouts per CDNA5 ISA 7.12.2:
//   A 16x4: VGPR0 = K=k0 (lanes 0-15) / K=k0+2 (lanes 16-31), VGPR1 = +1.
//   B  4x16 mirrors it; both become a float2 load from row (lane&15).
// ---------------------------------------------------------------------------
__device__ __forceinline__ void wmma_gemm_64xK(
    const float* __restrict__ A, int lda, int K,
    const float* __restrict__ W, int ldw,
    const float* __restrict__ bias,
    float* __restrict__ D, long ldd, int Nout,
    int n0, int lane) {
  const int  lh    = lane >> 4;          // half-wave select (K offset 0 / 2)
  const int  l16   = lane & 15;          // row / column within tile
  const bool colOK = (n0 + l16) < Nout;
  const float cmask = colOK ? 1.f : 0.f; // branchless tail mask -> EXEC stays all-1s
  const float* wrow = W + (size_t)(colOK ? (n0 + l16) : 0) * ldw;

  v8f acc0 = {}, acc1 = {}, acc2 = {}, acc3 = {};
#pragma unroll 2
  for (int k0 = 0; k0 < K; k0 += 4) {
    const int kk = k0 + 2 * lh;
    v2f bfrag = *(const v2f*)(wrow + kk);
    bfrag.x *= cmask;
    bfrag.y *= cmask;
    v2f a0 = *(const v2f*)(A + (size_t)(l16     ) * lda + kk);
    v2f a1 = *(const v2f*)(A + (size_t)(l16 + 16) * lda + kk);
    v2f a2 = *(const v2f*)(A + (size_t)(l16 + 32) * lda + kk);
    v2f a3 = *(const v2f*)(A + (size_t)(l16 + 48) * lda + kk);
    acc0 = __builtin_amdgcn_wmma_f32_16x16x4_f32(false, a0, false, bfrag, (short)0, acc0, false, false);
    acc1 = __builtin_amdgcn_wmma_f32_16x16x4_f32(false, a1, false, bfrag, (short)0, acc1, false, false);
    acc2 = __builtin_amdgcn_wmma_f32_16x16x4_f32(false, a2, false, bfrag, (short)0, acc2, false, false);
    acc3 = __builtin_amdgcn_wmma_f32_16x16x4_f32(false, a3, false, bfrag, (short)0, acc3, false, false);
  }

  if (colOK) {
    const long  nc = n0 + l16;
    const float bv = bias ? bias[nc] : 0.f;
    // C/D layout: VGPR i holds M=i (lanes 0-15) / M=8+i (lanes 16-31), N = lane&15
#pragma unroll
    for (int i = 0; i < 8; ++i) {
      D[(long)(lh * 8 + i     ) * ldd + nc] = acc0[i] + bv;
      D[(long)(lh * 8 + i + 16) * ldd + nc] = acc1[i] + bv;
      D[(long)(lh * 8 + i + 32) * ldd + nc] = acc2[i] + bv;
      D[(long)(lh * 8 + i + 48) * ldd + nc] = acc3[i] + bv;
    }
  }
}

// ---------------------------------------------------------------------------
// Kernel 2: GRU input/hidden gate GEMMs: gi = rnn @ W_ih^T + b_ih,
//                                        gh = h   @ W_hh^T + b_hh
// ---------------------------------------------------------------------------
__global__ void gru_gates_kernel(const float* __restrict__ rnn,
                                 const float* __restrict__ h,
                                 const float* __restrict__ W_ih,
                                 const float* __restrict__ W_hh,
                                 const float* __restrict__ b_ih,
                                 const float* __restrict__ b_hh,
                                 float* __restrict__ gi,
                                 float* __restrict__ gh) {
  int wave = blockIdx.x * (blockDim.x >> 5) + (threadIdx.x >> 5);
  int lane = threadIdx.x & 31;
  int n0 = wave * 16;
  if (n0 >= H3) return;
  wmma_gemm_64xK(rnn, KIN, KIN, W_ih, KIN, b_ih, gi, H3, H3, n0, lane);
  wmma_gemm_64xK(h,   H,   H,   W_hh, H,   b_hh, gh, H3, H3, n0, lane);
}

// ---------------------------------------------------------------------------
// Kernel 3: GRU gate combine (PyTorch order r, z, n); writes h_new to ws and
// to the second output slot.
// ---------------------------------------------------------------------------
__global__ void gru_combine_kernel(const float* __restrict__ gi,
                                   const float* __restrict__ gh,
                                   const float* __restrict__ h,
                                   float* __restrict__ h_new_ws,
                                   float* __restrict__ h_new_out) {
  int idx = blockIdx.x * blockDim.x + threadIdx.x;
  if (idx >= B * H) return;
  int b = idx / H;
  int j = idx - b * H;
  size_t base = (size_t)b * H3;
  float r = 1.f / (1.f + expf(-(gi[base + j]         + gh[base + j])));
  float z = 1.f / (1.f + expf(-(gi[base + H + j]     + gh[base + H + j])));
  float n = tanhf(gi[base + 2 * H + j] + r * gh[base + 2 * H + j]);
  float out = (1.f - z) * n + z * h[idx];
  h_new_ws[idx]  = out;
  h_new_out[idx] = out;
}

// ---------------------------------------------------------------------------
// Kernel 4: logits = h_new @ out_W^T + out_b   (64 x 50257, K = 1024)
// ---------------------------------------------------------------------------
__global__ void logits_gemm_kernel(const float* __restrict__ h_new,
                                   const float* __restrict__ out_W,
                                   const float* __restrict__ out_b,
                                   float* __restrict__ logits) {
  int wave = blockIdx.x * (blockDim.x >> 5) + (threadIdx.x >> 5);
  int lane = threadIdx.x & 31;
  int n0 = wave * 16;
  if (n0 >= V) return;
  wmma_gemm_64xK(h_new, H, H, out_W, H, out_b, logits, V, V, n0, lane);
}

// ---------------------------------------------------------------------------
// Kernel 5: row-wise log_softmax over V (one block per batch row)
// ---------------------------------------------------------------------------
__global__ void log_softmax_kernel(const float* __restrict__ logits,
                                   float* __restrict__ out) {
  const int b = blockIdx.x;
  const float* row = logits + (size_t)b * V;
  __shared__ float red[256];
  const int t = threadIdx.x;

  float m = -INFINITY;
  for (int v = t; v < V; v += 256) m = fmaxf(m, row[v]);
  red[t] = m;
  __syncthreads();
  for (int s = 128; s > 0; s >>= 1) {
    if (t < s) red[t] = fmaxf(red[t], red[t + s]);
    __syncthreads();
  }
  m = red[0];
  __syncthreads();

  float sum = 0.f;
  for (int v = t; v < V; v += 256) sum += expf(row[v] - m);
  red[t] = sum;
  __syncthreads();
  for (int s = 128; s > 0; s >>= 1) {
    if (t < s) red[t] += red[t + s];
    __syncthreads();
  }
  const float lse = m + logf(red[0]);

  float* orow = out + (size_t)b * V;
  for (int v = t; v < V; v += 256) orow[v] = row[v] - lse;
}

// ---------------------------------------------------------------------------
extern "C" void kernel_launch(void* const* d_in, const int* in_sizes, int n_in,
                              void* d_out, int out_size, void* d_ws, size_t ws_size,
                              hipStream_t stream) {
  const int*   tok   = (const int*)  d_in[0];   // decoder_input        (1,B)
  const float* h0    = (const float*)d_in[1];   // decoder_hidden_state (1,B,H)
  const float* enc   = (const float*)d_in[2];   // encoder_hidden_states(S,B,2H)
  const float* embt  = (const float*)d_in[3];   // emb_table            (V,E)
  // d_in[4] energy_W, d_in[5] energy_b: dead (softmax over singleton dim -> attn==1)
  const float* W_ih  = (const float*)d_in[6];   // (3H, 2H+E)
  const float* W_hh  = (const float*)d_in[7];   // (3H, H)
  const float* b_ih  = (const float*)d_in[8];
  const float* b_hh  = (const float*)d_in[9];
  const float* out_W = (const float*)d_in[10];  // (V, H)
  const float* out_b = (const float*)d_in[11];

  float* ws     = (float*)d_ws;
  float* rnn    = ws;                           // B*KIN   = 163840
  float* gi     = rnn   + (size_t)B * KIN;      // B*3H    = 196608
  float* gh     = gi    + (size_t)B * H3;       // B*3H    = 196608
  float* h_new  = gh    + (size_t)B * H3;       // B*H     =  65536
  float* logits = h_new + (size_t)B * H;        // B*V     = 3216448

  float* out_logp = (float*)d_out;              // (1,B,V)
  float* out_h    = out_logp + (size_t)B * V;   // (1,B,H)

  build_rnn_input_kernel<<<(B * KIN + 255) / 256, 256, 0, stream>>>(enc, tok, embt, rnn);

  gru_gates_kernel<<<(H3 / 16) / 4, 128, 0, stream>>>(rnn, h0, W_ih, W_hh, b_ih, b_hh, gi, gh);

  gru_combine_kernel<<<(B * H + 255) / 256, 256, 0, stream>>>(gi, gh, h0, h_new, out_h);

  const int ntiles = (V + 15) / 16;             // 3142
  logits_gemm_kernel<<<(ntiles + 3) / 4, 128, 0, stream>>>(h_new, out_W, out_b, logits);

  log_softmax_kernel<<<B, 256, 0, stream>>>(logits, out_logp);
}